// NBFNet_62852551409769
// MI455X (gfx1250) — compile-verified
//
#include <hip/hip_runtime.h>

#define N_ENT   40000
#define N_REL   401
#define D_DIM   64
#define L_LAYERS 6
#define B_DIM   4
#define ROWS    (N_ENT * B_DIM)   // 160000 rows of length 64

typedef __attribute__((ext_vector_type(2))) float v2f;
typedef __attribute__((ext_vector_type(8))) float v8f;

// ---------------------------------------------------------------- utilities
__global__ void zero_i32_kernel(int* __restrict__ p, int n) {
    int i = blockIdx.x * blockDim.x + threadIdx.x;
    if (i < n) p[i] = 0;
}
__global__ void zero_f32_kernel(float* __restrict__ p, int n) {
    int i = blockIdx.x * blockDim.x + threadIdx.x;
    if (i < n) p[i] = 0.0f;
}

// degree[t] = #edges with 1 <= r < 200 into t
__global__ void degree_kernel(const int* __restrict__ trip, int* __restrict__ deg, int E) {
    int e = blockIdx.x * blockDim.x + threadIdx.x;
    if (e >= E) return;
    int r = trip[e * 3 + 1];
    if (r >= 1 && r < (N_REL - 1) / 2)
        atomicAdd(&deg[trip[e * 3 + 2]], 1);
}

// x[n,b,d] = deg[n]>=3 ? ent_emb[n,d] : 1/sqrt(64)-0.5   (flat idx == n*256+b*64+d)
__global__ void init_kernel(const float* __restrict__ ent, const int* __restrict__ deg,
                            float* __restrict__ x) {
    int i = blockIdx.x * 256 + threadIdx.x;   // exactly N_ENT*256 threads
    int n = i >> 8;
    int d = i & 63;
    x[i] = (deg[n] >= 3) ? ent[n * D_DIM + d] : -0.375f;
}

// x[head[b], b, :] = q_emb[rel[b], :]
__global__ void headset_kernel(const int* __restrict__ head, const int* __restrict__ rel,
                               const float* __restrict__ q, float* __restrict__ x) {
    int t = threadIdx.x;          // 256 threads = B*D
    int b = t >> 6;
    int d = t & 63;
    int row = head[b] * B_DIM + b;
    x[row * D_DIM + d] = q[rel[b] * D_DIM + d];
}

// upd[t,b,d] += rel_w[r,d] * y[h,b,d]  over all edges; one thread per (edge,d)
__global__ void msg_kernel(const int* __restrict__ trip, const float* __restrict__ relw,
                           const float* __restrict__ y, float* __restrict__ upd, int E) {
    int i = blockIdx.x * blockDim.x + threadIdx.x;
    if (i >= E * D_DIM) return;
    int e = i >> 6;
    int d = i & 63;
    int h = trip[e * 3 + 0];
    int r = trip[e * 3 + 1];
    int t = trip[e * 3 + 2];
    float rw = relw[r * D_DIM + d];
    const float* ys = y + h * (B_DIM * D_DIM) + d;
    float* us = upd + t * (B_DIM * D_DIM) + d;
    #pragma unroll
    for (int b = 0; b < B_DIM; ++b)
        unsafeAtomicAdd(us + b * D_DIM, rw * ys[b * D_DIM]);
}

// out[M x 64] = in[M x 64] * W^T + bias (+resid) ; optional fused LayerNorm+ReLU.
// Workgroup: 256 threads = 8 wave32; 32 rows x 64 cols per WG.
// Wave w: mtile = w>>2 (16 rows), ntile = w&3 (16 cols); K=64 via 16x V_WMMA_F32_16X16X4_F32.
__global__ void __launch_bounds__(256) gemm_kernel(
    const float* __restrict__ in, const float* __restrict__ W,
    const float* __restrict__ bias, const float* __restrict__ resid,
    const float* __restrict__ g, const float* __restrict__ bb,
    float* __restrict__ out, int do_ln)
{
    __shared__ float sIn[32][68];   // padded: stride 68 breaks stride-64 bank conflicts
    __shared__ float sW[64][68];
    __shared__ float sOut[32][68];
    __shared__ float sMu[32];
    __shared__ float sRs[32];

    const int tid   = threadIdx.x;
    const int lane  = tid & 31;
    const int wave  = tid >> 5;
    const int mtile = wave >> 2;
    const int ntile = wave & 3;
    const int row_base = blockIdx.x * 32;

    // cooperative, coalesced staging of the 32x64 input tile and the full 64x64 W
    for (int j = tid; j < 64 * 64; j += 256) sW[j >> 6][j & 63] = W[j];
    for (int j = tid; j < 32 * 64; j += 256) sIn[j >> 6][j & 63] = in[row_base * 64 + j];
    __syncthreads();

    const int half = lane >> 4;        // K/M split across lane halves per ISA layout
    const int l15  = lane & 15;
    const int col  = ntile * 16 + l15; // N index for B and C/D fragments
    const int mrow = mtile * 16 + l15; // M index for A fragment

    // C/D layout: VGPR j, lanes 0-15 -> M=j, lanes 16-31 -> M=8+j, N=lane&15
    v8f acc;
    #pragma unroll
    for (int j = 0; j < 8; ++j) {
        int r = row_base + mtile * 16 + half * 8 + j;
        float c0 = bias[col];
        if (resid) c0 += resid[r * 64 + col];
        acc[j] = c0;
    }

    // A 16x4 f32: VGPR0 = {K=k0+2*half}, VGPR1 = {K=k0+2*half+1}, M = lane&15
    // B  4x16 f32: B[k][n] = W[n][k] -> contiguous pair from sW[col]
    #pragma unroll
    for (int k0 = 0; k0 < 64; k0 += 4) {
        int kk = k0 + half * 2;
        v2f a; a.x = sIn[mrow][kk]; a.y = sIn[mrow][kk + 1];
        v2f b; b.x = sW[col][kk];   b.y = sW[col][kk + 1];
        acc = __builtin_amdgcn_wmma_f32_16x16x4_f32(
            /*neg_a=*/false, a, /*neg_b=*/false, b,
            /*c_mod=*/(short)0, acc, /*reuse_a=*/false, /*reuse_b=*/false);
    }

    if (!do_ln) {
        #pragma unroll
        for (int j = 0; j < 8; ++j) {
            int r = row_base + mtile * 16 + half * 8 + j;
            out[r * 64 + col] = acc[j];
        }
        return;
    }

    // fused LayerNorm + ReLU epilogue (stage tile in LDS, per-row stats, coalesced store)
    #pragma unroll
    for (int j = 0; j < 8; ++j)
        sOut[mtile * 16 + half * 8 + j][col] = acc[j];
    __syncthreads();

    if (tid < 32) {
        float s = 0.f, ss = 0.f;
        #pragma unroll
        for (int c = 0; c < 64; ++c) { float v = sOut[tid][c]; s += v; ss += v * v; }
        float mu  = s * (1.0f / 64.0f);
        float var = ss * (1.0f / 64.0f) - mu * mu;     // population variance (ddof=0)
        sMu[tid] = mu;
        sRs[tid] = rsqrtf(var + 1e-5f);
    }
    __syncthreads();

    for (int j = tid; j < 32 * 64; j += 256) {
        int rr = j >> 6, cc = j & 63;
        float v = (sOut[rr][cc] - sMu[rr]) * sRs[rr] * g[cc] + bb[cc];
        out[row_base * 64 + j] = fmaxf(v, 0.0f);
    }
}

// out[b, n] = dot(x[n,b,:], W_cls) + b_cls ; one wave32 per row, shuffle reduction
__global__ void cls_kernel(const float* __restrict__ x, const float* __restrict__ Wc,
                           const float* __restrict__ bc, float* __restrict__ out) {
    int row  = blockIdx.x * 8 + (threadIdx.x >> 5);   // 8 waves per block
    int lane = threadIdx.x & 31;
    const float* xr = x + row * D_DIM;
    float s = xr[lane] * Wc[lane] + xr[lane + 32] * Wc[lane + 32];
    #pragma unroll
    for (int off = 16; off > 0; off >>= 1) s += __shfl_xor(s, off, 32);
    if (lane == 0) {
        int n = row >> 2;     // row = n*B + b
        int b = row & 3;
        out[b * N_ENT + n] = s + bc[0];
    }
}

// ---------------------------------------------------------------- launcher
extern "C" void kernel_launch(void* const* d_in, const int* in_sizes, int n_in,
                              void* d_out, int out_size, void* d_ws, size_t ws_size,
                              hipStream_t stream) {
    const int*   head    = (const int*)  d_in[0];
    const int*   rel     = (const int*)  d_in[1];
    const int*   triples = (const int*)  d_in[2];
    const float* ent_emb = (const float*)d_in[3];
    const float* q_emb   = (const float*)d_in[4];
    const float* rel_w   = (const float*)d_in[5];
    const float* W_ent   = (const float*)d_in[6];
    const float* b_ent   = (const float*)d_in[7];
    const float* W_lin   = (const float*)d_in[8];
    const float* b_lin   = (const float*)d_in[9];
    const float* ln_g    = (const float*)d_in[10];
    const float* ln_b    = (const float*)d_in[11];
    const float* W_cls   = (const float*)d_in[12];
    const float* b_cls   = (const float*)d_in[13];
    float* out = (float*)d_out;

    const int E = in_sizes[2] / 3;
    const int n_state = ROWS * D_DIM;             // 10,240,000 floats

    float* x   = (float*)d_ws;
    float* y   = x + n_state;
    float* upd = y + n_state;
    int*   deg = (int*)(upd + n_state);

    // degree + const_mask + init
    zero_i32_kernel<<<(N_ENT + 255) / 256, 256, 0, stream>>>(deg, N_ENT);
    degree_kernel<<<(E + 255) / 256, 256, 0, stream>>>(triples, deg, E);
    init_kernel<<<N_ENT, 256, 0, stream>>>(ent_emb, deg, x);
    headset_kernel<<<1, 256, 0, stream>>>(head, rel, q_emb, x);

    const int gemm_blocks = ROWS / 32;            // 5000
    const int msg_blocks  = (E * D_DIM + 255) / 256;

    for (int i = 0; i < L_LAYERS; ++i) {
        // y = x * W_ent[i]^T + b_ent[i]
        gemm_kernel<<<gemm_blocks, 256, 0, stream>>>(
            x, W_ent + i * D_DIM * D_DIM, b_ent + i * D_DIM,
            nullptr, nullptr, nullptr, y, 0);
        // upd = segment_sum over edges of rel_w[i][r] * y[h]
        zero_f32_kernel<<<(n_state + 255) / 256, 256, 0, stream>>>(upd, n_state);
        msg_kernel<<<msg_blocks, 256, 0, stream>>>(
            triples, rel_w + i * N_REL * D_DIM, y, upd, E);
        // x = relu(layernorm(upd * W_lin[i]^T + b_lin[i] + y))
        gemm_kernel<<<gemm_blocks, 256, 0, stream>>>(
            upd, W_lin + i * D_DIM * D_DIM, b_lin + i * D_DIM,
            y, ln_g + i * D_DIM, ln_b + i * D_DIM, x, 1);
    }

    cls_kernel<<<ROWS / 8, 256, 0, stream>>>(x, W_cls, b_cls, out);
}